// DenseRagged_34832184771173
// MI455X (gfx1250) — compile-verified
//
#include <hip/hip_runtime.h>
#include <hip/hip_bf16.h>

typedef float v2f __attribute__((ext_vector_type(2)));
typedef float v4f __attribute__((ext_vector_type(4)));
typedef float v8f __attribute__((ext_vector_type(8)));
typedef int   v4i __attribute__((ext_vector_type(4)));

typedef __attribute__((address_space(1))) v4i* gptr_v4i;   // global int4*
typedef __attribute__((address_space(3))) v4i* lptr_v4i;   // LDS int4*

// Padded LDS row stride (floats): bank index advances by 4 per row (mod 64),
// so a half-wave's 16 rows hit 16 distinct banks; the two half-waves (k offset
// by 2 floats) land in disjoint mod-4 bank classes.
#define LDS_STRIDE   68
#define TILE_ROWS    64
#define TILES_PER_WG 4

#if defined(__has_builtin)
#if __has_builtin(__builtin_amdgcn_global_load_async_to_lds_b128)
#define ASYNC_STAGE 1
#endif
#endif
#ifndef ASYNC_STAGE
#define ASYNC_STAGE 0
#endif

// Stage one 64x64 fp32 x-tile (row-padded) into LDS.
// 64 rows * 16 float4 = 1024 xfers; 256 threads -> 4 passes, fully coalesced.
__device__ __forceinline__ void stage_tile(const float* __restrict__ srcx,
                                           float* dst, int tid)
{
#pragma unroll
    for (int p = 0; p < 4; ++p) {
        int idx = p * 256 + tid;
        int row = idx >> 4;          // 0..63
        int q   = idx & 15;          // float4 within row
#if ASYNC_STAGE
        // CDNA5 async DMA: global -> LDS without touching VGPRs (ASYNCcnt).
        __builtin_amdgcn_global_load_async_to_lds_b128(
            (gptr_v4i)(__attribute__((address_space(1))) void*)(srcx + row * 64 + q * 4),
            (lptr_v4i)(__attribute__((address_space(3))) void*)(dst + row * LDS_STRIDE + q * 4),
            0, 0);
#else
        v4f v = *(const v4f*)(srcx + row * 64 + q * 4);
        *(v4f*)(dst + row * LDS_STRIDE + q * 4) = v;
#endif
    }
}

__device__ __forceinline__ void wait_stage()
{
#if ASYNC_STAGE
#if __has_builtin(__builtin_amdgcn_s_wait_asynccnt)
    __builtin_amdgcn_s_wait_asynccnt(0);
#else
    asm volatile("s_wait_asynccnt 0" ::: "memory");
#endif
#endif
}

// C = relu(x[M,64] @ W[64,128] + b[128])
// Workgroup: 256 threads = 8 waves; processes TILES_PER_WG consecutive
// 64-row M-tiles with a double-buffered LDS pipeline. Wave w owns N tile
// [16w,16w+16); per tile: 4 M-subtiles x 16 K-steps of V_WMMA_F32_16X16X4_F32.
__global__ __launch_bounds__(256) void dense_bias_relu_f32_wmma(
    const float* __restrict__ x,     // [M, 64]
    const float* __restrict__ Wm,    // [64, 128]
    const float* __restrict__ bias,  // [128]
    float* __restrict__ out)         // [M, 128]
{
    __shared__ float ldsx[2][TILE_ROWS * LDS_STRIDE];   // ~34 KB

    const int tid  = threadIdx.x;
    const int lane = tid & 31;
    const int wave = tid >> 5;          // 0..7 -> N tile
    const int half = lane >> 4;         // 0/1 half-wave
    const int l15  = lane & 15;
    const long tile0 = (long)blockIdx.x * TILES_PER_WG;

    // Kick off stage of the first tile ASAP.
    stage_tile(x + tile0 * TILE_ROWS * 64, ldsx[0], tid);

    // ---- B fragments + bias: loaded ONCE per workgroup (amortized over
    // TILES_PER_WG tiles). B layout for 16x16x4 f32 WMMA (4x16 tile, 2 VGPRs):
    //   VGPR j, lanes 0-15:  K = j,     N = lane
    //   VGPR j, lanes 16-31: K = j + 2, N = lane - 16
    const int nb = wave * 16 + l15;
    v2f Bfrag[16];
#pragma unroll
    for (int k = 0; k < 16; ++k) {
        int kr = 4 * k + 2 * half;
        Bfrag[k].x = Wm[(kr + 0) * 128 + nb];
        Bfrag[k].y = Wm[(kr + 1) * 128 + nb];
    }
    const float bv = bias[nb];          // same N column for all 8 C elements

    for (int t = 0; t < TILES_PER_WG; ++t) {
        wait_stage();
        __syncthreads();   // publish staged buf[t&1]; also fences compute of
                           // t-1 (which read buf[(t+1)&1]) before re-staging it

        if (t + 1 < TILES_PER_WG)
            stage_tile(x + (tile0 + t + 1) * TILE_ROWS * 64,
                       ldsx[(t + 1) & 1], tid);

        const float* buf = ldsx[t & 1];
        const long   m0  = (tile0 + t) * TILE_ROWS;

        // A layout (16x4 tile, 2 VGPRs):
        //   VGPR j, lanes 0-15:  M = lane, K = j
        //   VGPR j, lanes 16-31: M = lane - 16, K = j + 2
        // => lane reads x[mrow][4k + 2*half .. +1] as one ds_load_b64.
#pragma unroll
        for (int msub = 0; msub < 4; ++msub) {
            v8f acc = {0.f, 0.f, 0.f, 0.f, 0.f, 0.f, 0.f, 0.f};
            const float* arow = &buf[(msub * 16 + l15) * LDS_STRIDE + 2 * half];
#pragma unroll
            for (int k = 0; k < 16; ++k) {
                v2f a = *(const v2f*)(arow + 4 * k);
                acc = __builtin_amdgcn_wmma_f32_16x16x4_f32(
                    /*neg_a=*/false, a, /*neg_b=*/false, Bfrag[k],
                    /*c_mod=*/(short)0, acc,
                    /*reuse_a=*/false, /*reuse_b=*/false);
            }

            // Epilogue: bias + relu + store.
            // C/D layout: VGPR i -> M = msub*16 + i + 8*half, N = wave*16+l15.
            const long rbase = m0 + msub * 16 + half * 8;
            float* o = out + rbase * 128 + wave * 16 + l15;
#pragma unroll
            for (int i = 0; i < 8; ++i) {
                float r = acc[i] + bv;
                o[(long)i * 128] = r > 0.f ? r : 0.f;
            }
        }
    }
}

extern "C" void kernel_launch(void* const* d_in, const int* in_sizes, int n_in,
                              void* d_out, int out_size, void* d_ws, size_t ws_size,
                              hipStream_t stream) {
    const float* x  = (const float*)d_in[0];   // [M,64] fp32
    const float* Wm = (const float*)d_in[1];   // [64,128] fp32
    const float* b  = (const float*)d_in[2];   // [128] fp32
    float* out = (float*)d_out;                // [M,128] fp32

    const long M = (long)in_sizes[0] / 64;             // 4,000,000 rows
    const int grid = (int)(M / (TILE_ROWS * TILES_PER_WG));  // 15,625 blocks

    dense_bias_relu_f32_wmma<<<grid, 256, 0, stream>>>(x, Wm, b, out);
}